// DiscriminativeLoss_20822001451156
// MI455X (gfx1250) — compile-verified
//
#include <hip/hip_runtime.h>

// DiscriminativeLoss for MI455X (gfx1250, wave32).
//
// Roofline: 2 streaming passes over 144 MiB => ~288 MiB traffic => ~12us at
// 23.3 TB/s. Compute is ~50 VALU ops / 36 bytes => memory bound. The 5x8
// center Gram matrix per batch (pairwise distance loss) maps onto
// V_WMMA_F32_16X16X4_F32 (two K=4 slabs of D=8); symmetric Gram lets A and B
// share one operand register pair under the ISA 16x4 f32 layout.

typedef __attribute__((ext_vector_type(2))) float v2f;
typedef __attribute__((ext_vector_type(8))) float v8f;

#define K_INST 5
#define D_EMB 8
#define P_IMG (512 * 1024)
#define B_IMG 8
#define WS_ACC_STRIDE 48                 // per batch: 40 sums + 5 counts + pad
#define WS_VAL_OFF (B_IMG * WS_ACC_STRIDE)  // 384
#define WS_TOTAL (WS_VAL_OFF + B_IMG * 8)   // 448 floats of workspace
#define BLOCKS_PER_B 128
#define DELTA_V 0.5f
#define TWO_DELTA_D 6.0f

__device__ __forceinline__ float waveReduceSum(float v) {
#pragma unroll
  for (int off = 16; off > 0; off >>= 1) v += __shfl_xor(v, off, 32);
  return v;
}

__global__ void zero_ws(float* __restrict__ ws, float* __restrict__ out) {
  const int t = blockIdx.x * blockDim.x + threadIdx.x;
  if (t < WS_TOTAL) ws[t] = 0.f;
  if (t < 4) out[t] = 0.f;
}

// Pass 1: per-(b,k) counts and per-dim sums. float4 loads => 512B/wave/instr.
__global__ __launch_bounds__(256) void accum_sums(const float* __restrict__ emb,
                                                  const int* __restrict__ mask,
                                                  float* __restrict__ ws_acc) {
  const int b = blockIdx.y;
  const int tid = threadIdx.x;
  float cnt[K_INST];
  float sum[K_INST][D_EMB];
#pragma unroll
  for (int k = 0; k < K_INST; ++k) {
    cnt[k] = 0.f;
#pragma unroll
    for (int d = 0; d < D_EMB; ++d) sum[k][d] = 0.f;
  }
  const float* embB = emb + (size_t)b * D_EMB * P_IMG;
  const int* maskB = mask + (size_t)b * P_IMG;
  const int stride4 = gridDim.x * blockDim.x * 4;
  for (int p4 = (blockIdx.x * blockDim.x + tid) * 4; p4 < P_IMG; p4 += stride4) {
    const int4 lab4 = *(const int4*)&maskB[p4];
    float4 e4[D_EMB];
#pragma unroll
    for (int d = 0; d < D_EMB; ++d) e4[d] = *(const float4*)&embB[d * P_IMG + p4];
    const int labs[4] = {lab4.x, lab4.y, lab4.z, lab4.w};
#pragma unroll
    for (int s = 0; s < 4; ++s) {
#pragma unroll
      for (int k = 0; k < K_INST; ++k) {
        const float m = (labs[s] == k + 1) ? 1.f : 0.f;
        cnt[k] += m;
#pragma unroll
        for (int d = 0; d < D_EMB; ++d)
          sum[k][d] = fmaf(m, (&e4[d].x)[s], sum[k][d]);
      }
    }
  }
  __shared__ float s_acc[45];
  if (tid < 45) s_acc[tid] = 0.f;
  __syncthreads();
  const int lane = tid & 31;
#pragma unroll
  for (int k = 0; k < K_INST; ++k) {
#pragma unroll
    for (int d = 0; d < D_EMB; ++d) {
      const float r = waveReduceSum(sum[k][d]);
      if (lane == 0) atomicAdd(&s_acc[k * D_EMB + d], r);
    }
    const float rc = waveReduceSum(cnt[k]);
    if (lane == 0) atomicAdd(&s_acc[40 + k], rc);
  }
  __syncthreads();
  if (tid < 45) atomicAdd(&ws_acc[b * WS_ACC_STRIDE + tid], s_acc[tid]);
}

// Pass 2: per-pixel hinge^2 of distance to own center, reduced per (b,k).
__global__ __launch_bounds__(256) void var_loss(const float* __restrict__ emb,
                                                const int* __restrict__ mask,
                                                const float* __restrict__ ws_acc,
                                                float* __restrict__ ws_val) {
  const int b = blockIdx.y;
  const int tid = threadIdx.x;
  __shared__ float s_c[40];
  __shared__ float s_v[K_INST];
  if (tid < 40) {
    const int k = tid >> 3;
    const float c = ws_acc[b * WS_ACC_STRIDE + 40 + k];
    s_c[tid] = ws_acc[b * WS_ACC_STRIDE + tid] / fmaxf(c, 1.f);
  }
  if (tid < K_INST) s_v[tid] = 0.f;
  __syncthreads();
  float acc[K_INST] = {0.f, 0.f, 0.f, 0.f, 0.f};
  const float* embB = emb + (size_t)b * D_EMB * P_IMG;
  const int* maskB = mask + (size_t)b * P_IMG;
  const int stride4 = gridDim.x * blockDim.x * 4;
  for (int p4 = (blockIdx.x * blockDim.x + tid) * 4; p4 < P_IMG; p4 += stride4) {
    const int4 lab4 = *(const int4*)&maskB[p4];
    float4 e4[D_EMB];
#pragma unroll
    for (int d = 0; d < D_EMB; ++d) e4[d] = *(const float4*)&embB[d * P_IMG + p4];
    const int labs[4] = {lab4.x, lab4.y, lab4.z, lab4.w};
#pragma unroll
    for (int s = 0; s < 4; ++s) {
      const int lab = labs[s];
      const int k = min(max(lab - 1, 0), K_INST - 1);
      float d2 = 0.f;
#pragma unroll
      for (int d = 0; d < D_EMB; ++d) {
        const float diff = (&e4[d].x)[s] - s_c[k * D_EMB + d];
        d2 = fmaf(diff, diff, d2);
      }
      const float hv = fmaxf(sqrtf(d2) - DELTA_V, 0.f);
      const float val = hv * hv;
#pragma unroll
      for (int kk = 0; kk < K_INST; ++kk)
        acc[kk] += (lab == kk + 1) ? val : 0.f;
    }
  }
  const int lane = tid & 31;
#pragma unroll
  for (int k = 0; k < K_INST; ++k) {
    const float r = waveReduceSum(acc[k]);
    if (lane == 0) atomicAdd(&s_v[k], r);
  }
  __syncthreads();
  if (tid < K_INST) atomicAdd(&ws_val[b * 8 + tid], s_v[tid]);
}

// Epilogue: one block, 8 waves; wave w handles batch w. Center Gram via WMMA.
__global__ __launch_bounds__(256) void finalize(const float* __restrict__ ws_acc,
                                                const float* __restrict__ ws_val,
                                                float* __restrict__ out) {
  __shared__ float s_cent[B_IMG][56];  // [k*8+d] centers; [40+k] counts; [48+k] |c|^2
  __shared__ float s_pair[B_IMG][K_INST];
  __shared__ float s_varp[B_IMG][K_INST];
  __shared__ float s_regp[B_IMG][K_INST];
  __shared__ float s_res[B_IMG][4];
  const int tid = threadIdx.x;
  // phase 1: centers + counts for all batches (block-wide)
  for (int i = tid; i < B_IMG * 40; i += 256) {
    const int b = i / 40, j = i % 40, k = j >> 3;
    const float c = ws_acc[b * WS_ACC_STRIDE + 40 + k];
    s_cent[b][j] = ws_acc[b * WS_ACC_STRIDE + j] / fmaxf(c, 1.f);
  }
  for (int i = tid; i < B_IMG * K_INST; i += 256) {
    const int b = i / K_INST, k = i % K_INST;
    s_cent[b][40 + k] = ws_acc[b * WS_ACC_STRIDE + 40 + k];
  }
  __syncthreads();
  const int w = tid >> 5;  // wave == batch
  const int lane = tid & 31;
  // phase 2: center norms
  if (lane < K_INST) {
    float ns = 0.f;
#pragma unroll
    for (int d = 0; d < D_EMB; ++d) {
      const float c = s_cent[w][lane * D_EMB + d];
      ns = fmaf(c, c, ns);
    }
    s_cent[w][48 + lane] = ns;
  }
  __syncthreads();
  // phase 3: Gram = C * C^T via V_WMMA_F32_16X16X4_F32 (full EXEC here).
  // 32-bit A 16x4 layout: lanes 0-15 hold K={0,1} (VGPR0/1), lanes 16-31 K={2,3};
  // B follows the same half-wave K split with N across lanes, so the symmetric
  // Gram uses one operand pair for both A and B. Absent rows are zero.
  v8f g = {};
  const int row = lane & 15;
  const int half = lane >> 4;
#pragma unroll
  for (int c0 = 0; c0 < D_EMB; c0 += 4) {
    const int kb = c0 + (half << 1);
    v2f a;
    a.x = (row < K_INST) ? s_cent[w][row * D_EMB + kb] : 0.f;
    a.y = (row < K_INST) ? s_cent[w][row * D_EMB + kb + 1] : 0.f;
    g = __builtin_amdgcn_wmma_f32_16x16x4_f32(false, a, false, a, (short)0, g,
                                              false, false);
  }
  // g[i] at lane j (j<16) == G[i][j] = c_i . c_j
  if (lane < K_INST) {
    const float cntk = s_cent[w][40 + lane];
    const float pj = (cntk > 0.f) ? 1.f : 0.f;
    const float nsj = s_cent[w][48 + lane];
    float pairsum = 0.f;
#pragma unroll
    for (int i = 0; i < K_INST; ++i) {
      const float pi = (s_cent[w][40 + i] > 0.f) ? 1.f : 0.f;
      const float ds2 = fmaxf(s_cent[w][48 + i] + nsj - 2.f * g[i], 0.f);
      const float hv = fmaxf(TWO_DELTA_D - sqrtf(ds2), 0.f);
      const float term = hv * hv * pi * pj;
      pairsum += (i < lane) ? term : 0.f;  // i<j upper triangle
    }
    s_pair[w][lane] = pairsum;
    s_varp[w][lane] = (ws_val[w * 8 + lane] / fmaxf(cntk, 1.f)) * pj;
    s_regp[w][lane] = sqrtf(nsj) * pj;
  }
  __syncthreads();
  // phase 4: per-batch losses
  if (tid < B_IMG) {
    const int b = tid;
    float N = 0.f, pv = 0.f, pd = 0.f, pr = 0.f;
#pragma unroll
    for (int k = 0; k < K_INST; ++k) {
      N += (s_cent[b][40 + k] > 0.f) ? 1.f : 0.f;
      pv += s_varp[b][k];
      pd += s_pair[b][k];
      pr += s_regp[b][k];
    }
    const float Nc = fmaxf(N, 1.f);
    const float np = N * (N - 1.f) * 0.5f;
    s_res[b][0] = pv / Nc;
    s_res[b][1] = pd / ((N > 1.f) ? np : 1.f);
    s_res[b][2] = pr / Nc;
    s_res[b][3] = (N > 0.f) ? 1.f : 0.f;
  }
  __syncthreads();
  // phase 5: batch mean + total
  if (tid == 0) {
    float sv = 0.f, sd = 0.f, sr = 0.f, sh = 0.f;
#pragma unroll
    for (int b = 0; b < B_IMG; ++b) {
      const float h = s_res[b][3];
      sv += s_res[b][0] * h;
      sd += s_res[b][1] * h;
      sr += s_res[b][2] * h;
      sh += h;
    }
    const float denom = fmaxf(sh, 1.f);
    const float lv = sv / denom, ld = sd / denom, lr = sr / denom;
    out[1] = lv;
    out[2] = ld;
    out[3] = lr;
    out[0] = lv + ld + 0.001f * lr;  // ALPHA=BETA=1, GAMMA=0.001
  }
}

extern "C" void kernel_launch(void* const* d_in, const int* in_sizes, int n_in,
                              void* d_out, int out_size, void* d_ws,
                              size_t ws_size, hipStream_t stream) {
  (void)in_sizes; (void)n_in; (void)out_size; (void)ws_size;
  const float* emb = (const float*)d_in[0];
  const int* mask = (const int*)d_in[1];
  float* out = (float*)d_out;
  float* ws = (float*)d_ws;
  float* ws_acc = ws;
  float* ws_val = ws + WS_VAL_OFF;

  zero_ws<<<1, 512, 0, stream>>>(ws, out);
  dim3 grid(BLOCKS_PER_B, B_IMG);
  accum_sums<<<grid, 256, 0, stream>>>(emb, mask, ws_acc);
  var_loss<<<grid, 256, 0, stream>>>(emb, mask, ws_acc, ws_val);
  finalize<<<1, 256, 0, stream>>>(ws_acc, ws_val, out);
}